// GaussianKernelSimilarity_71502615544486
// MI455X (gfx1250) — compile-verified
//
#include <hip/hip_runtime.h>

typedef __attribute__((ext_vector_type(2))) float v2f;
typedef __attribute__((ext_vector_type(8))) float v8f;

#define N_TOK 8192
#define N_EXP 4096
#define DIM   1024
#define LDA   20      // padded LDS row stride (floats): 80B, 16B-aligned, conflict-free
#define LDB2  132     // padded K-major B stride for GEMM2: 528B, 16B-aligned

#define WMMA_F32(A, B, C) \
  __builtin_amdgcn_wmma_f32_16x16x4_f32(false, (A), false, (B), (short)0, (C), false, false)

// ---------------- kernel 1: row sums of squares for z and expert_keys ----------------
__global__ __launch_bounds__(256)
void rowsq_kernel(const float* __restrict__ z, const float* __restrict__ e,
                  float* __restrict__ zsq, float* __restrict__ esq) {
  __shared__ float red[256];
  const int r = blockIdx.x;
  const float* src;
  float* dst;
  if (r < N_TOK) { src = z + (size_t)r * DIM;           dst = zsq + r; }
  else           { src = e + (size_t)(r - N_TOK) * DIM; dst = esq + (r - N_TOK); }
  float s = 0.f;
  for (int i = threadIdx.x; i < DIM; i += 256) { float v = src[i]; s += v * v; }
  red[threadIdx.x] = s;
  __syncthreads();
  for (int off = 128; off > 0; off >>= 1) {
    if (threadIdx.x < off) red[threadIdx.x] += red[threadIdx.x + off];
    __syncthreads();
  }
  if (threadIdx.x == 0) *dst = red[0];
}

// ---------------- kernel 2: sim = exp(-dist2/2), fp32 WMMA GEMM ----------------
// Block tile 128(M) x 64(N), 8 waves, each wave a 32x32 tile (4 accumulators,
// every A/B fragment feeds 2 WMMAs -> 16 WMMAs per 16-K chunk vs 16 frag loads).
__global__ __launch_bounds__(256)
void sim_gemm_kernel(const float* __restrict__ z, const float* __restrict__ e,
                     const float* __restrict__ zsq, const float* __restrict__ esq,
                     float* __restrict__ sim) {
  __shared__ float lA[128 * LDA];
  __shared__ float lB[64 * LDA];
  const int tid  = threadIdx.x;
  const int wave = tid >> 5;
  const int lane = tid & 31;
  const int l16  = lane & 15;
  const int koff = (lane >> 4) << 1;          // lanes 16-31 hold K+2 (ISA f32 layout)
  const int m0 = blockIdx.y * 128;
  const int n0 = blockIdx.x * 64;
  const int mw = (wave & 3) * 32;             // wave's M offset in block tile
  const int nw = (wave >> 2) * 32;            // wave's N offset in block tile
  const int lrow = tid >> 2;                  // 0..63 cooperative fill row
  const int lkq  = (tid & 3) << 2;            // 0,4,8,12 float4 chunk

  v8f acc00 = {}, acc01 = {}, acc10 = {}, acc11 = {};

  for (int k0 = 0; k0 < DIM; k0 += 16) {
    const float4 a40 = *(const float4*)(z + (size_t)(m0 + lrow) * DIM + k0 + lkq);
    const float4 a41 = *(const float4*)(z + (size_t)(m0 + 64 + lrow) * DIM + k0 + lkq);
    const float4 b4  = *(const float4*)(e + (size_t)(n0 + lrow) * DIM + k0 + lkq);
    __syncthreads();                          // previous chunk's fragment reads done
    *(float4*)(lA + lrow * LDA + lkq)        = a40;
    *(float4*)(lA + (64 + lrow) * LDA + lkq) = a41;
    *(float4*)(lB + lrow * LDA + lkq)        = b4;
    __syncthreads();
    const float* pa0 = lA + (mw + l16) * LDA + koff;
    const float* pa1 = lA + (mw + 16 + l16) * LDA + koff;
    const float* pb0 = lB + (nw + l16) * LDA + koff;
    const float* pb1 = lB + (nw + 16 + l16) * LDA + koff;
#pragma unroll
    for (int kk = 0; kk < 16; kk += 4) {
      v2f a0 = { pa0[kk], pa0[kk + 1] };
      v2f a1 = { pa1[kk], pa1[kk + 1] };
      v2f b0 = { pb0[kk], pb0[kk + 1] };      // B[k][n] = e[n][k]
      v2f b1 = { pb1[kk], pb1[kk + 1] };
      acc00 = WMMA_F32(a0, b0, acc00);
      acc01 = WMMA_F32(a0, b1, acc01);
      acc10 = WMMA_F32(a1, b0, acc10);
      acc11 = WMMA_F32(a1, b1, acc11);
    }
  }
  // C/D layout: element i at lane L -> M = i + 8*(L/16), N = L%16
  const int mb0 = m0 + mw + ((lane >> 4) << 3);
  const int mb1 = mb0 + 16;
  const int nb0 = n0 + nw + l16;
  const int nb1 = nb0 + 16;
  const float eq0 = esq[nb0];
  const float eq1 = esq[nb1];
#pragma unroll
  for (int i = 0; i < 8; ++i) {
    const int ma = mb0 + i;
    const int mbm = mb1 + i;
    const float zqa = zsq[ma];
    const float zqb = zsq[mbm];
    float d00 = fmaxf(zqa + eq0 - 2.0f * acc00[i], 0.0f);
    float d01 = fmaxf(zqa + eq1 - 2.0f * acc01[i], 0.0f);
    float d10 = fmaxf(zqb + eq0 - 2.0f * acc10[i], 0.0f);
    float d11 = fmaxf(zqb + eq1 - 2.0f * acc11[i], 0.0f);
    sim[(size_t)ma  * N_EXP + nb0] = __expf(-0.5f * d00);
    sim[(size_t)ma  * N_EXP + nb1] = __expf(-0.5f * d01);
    sim[(size_t)mbm * N_EXP + nb0] = __expf(-0.5f * d10);
    sim[(size_t)mbm * N_EXP + nb1] = __expf(-0.5f * d11);
  }
}

// ---------------- kernel 3: inv_rowsum[n] = 1 / sum_e exp(sim[n,e]) ----------------
__global__ __launch_bounds__(256)
void rowsum_kernel(const float* __restrict__ sim, float* __restrict__ invrow) {
  __shared__ float red[256];
  const float* row = sim + (size_t)blockIdx.x * N_EXP;
  float s = 0.f;
  for (int i = threadIdx.x; i < N_EXP; i += 256) s += __expf(row[i]);
  red[threadIdx.x] = s;
  __syncthreads();
  for (int off = 128; off > 0; off >>= 1) {
    if (threadIdx.x < off) red[threadIdx.x] += red[threadIdx.x + off];
    __syncthreads();
  }
  if (threadIdx.x == 0) invrow[blockIdx.x] = 1.0f / red[0];
}

// ---------------- kernel 4: weighted = softmax(sim) @ expert_keys ----------------
// Block tile 32(M) x 128(N), 8 waves, each wave 32x16 (2 accumulators share the
// B fragment); softmax weights materialized into LDS once per block tile.
__global__ __launch_bounds__(256)
void weight_gemm_kernel(const float* __restrict__ sim, const float* __restrict__ e,
                        const float* __restrict__ invrow, float* __restrict__ out) {
  __shared__ float lA[32 * LDA];              // weights tile  [M=32][K=16]
  __shared__ float lB[16 * LDB2];             // key tile      [K=16][N=128]
  const int tid  = threadIdx.x;
  const int wave = tid >> 5;
  const int lane = tid & 31;
  const int l16  = lane & 15;
  const int koff = (lane >> 4) << 1;
  const int m0 = blockIdx.y * 32;
  const int n0 = blockIdx.x * 128;
  const int am = tid >> 4;                    // 0..15 : A fill rows am, am+16
  const int ak = tid & 15;
  const int bk = tid >> 5;                    // 0..7  : B fill rows bk, bk+8
  const int bn = (tid & 31) << 2;             // float4 column
  const float inv0 = invrow[m0 + am];
  const float inv1 = invrow[m0 + 16 + am];

  v8f acc0 = {}, acc1 = {};
  for (int k0 = 0; k0 < N_EXP; k0 += 16) {
    const float s0 = sim[(size_t)(m0 + am) * N_EXP + k0 + ak];
    const float s1 = sim[(size_t)(m0 + 16 + am) * N_EXP + k0 + ak];
    const float4 b0 = *(const float4*)(e + (size_t)(k0 + bk) * DIM + n0 + bn);
    const float4 b1 = *(const float4*)(e + (size_t)(k0 + 8 + bk) * DIM + n0 + bn);
    __syncthreads();
    lA[am * LDA + ak]        = __expf(s0) * inv0;
    lA[(16 + am) * LDA + ak] = __expf(s1) * inv1;
    *(float4*)(lB + bk * LDB2 + bn)       = b0;
    *(float4*)(lB + (8 + bk) * LDB2 + bn) = b1;
    __syncthreads();
    const float* pa0 = lA + l16 * LDA + koff;
    const float* pa1 = lA + (16 + l16) * LDA + koff;
    const float* pb  = lB + koff * LDB2 + wave * 16 + l16;
#pragma unroll
    for (int kk = 0; kk < 16; kk += 4) {
      v2f a0 = { pa0[kk], pa0[kk + 1] };
      v2f a1 = { pa1[kk], pa1[kk + 1] };
      v2f b  = { pb[kk * LDB2], pb[(kk + 1) * LDB2] };
      acc0 = WMMA_F32(a0, b, acc0);
      acc1 = WMMA_F32(a1, b, acc1);
    }
  }
  const int mb = m0 + ((lane >> 4) << 3);
  const int n  = n0 + wave * 16 + l16;
#pragma unroll
  for (int i = 0; i < 8; ++i) {
    out[(size_t)(mb + i) * DIM + n]      = acc0[i];
    out[(size_t)(mb + 16 + i) * DIM + n] = acc1[i];
  }
}

// ---------------- launcher ----------------
extern "C" void kernel_launch(void* const* d_in, const int* in_sizes, int n_in,
                              void* d_out, int out_size, void* d_ws, size_t ws_size,
                              hipStream_t stream) {
  (void)in_sizes; (void)n_in; (void)out_size; (void)ws_size;
  const float* z = (const float*)d_in[0];
  const float* e = (const float*)d_in[1];
  float* sim = (float*)d_out;                              // [8192, 4096]
  float* we  = (float*)d_out + (size_t)N_TOK * N_EXP;      // [8192, 1024]
  float* zsq    = (float*)d_ws;                            // 8192
  float* esq    = zsq + N_TOK;                             // 4096
  float* invrow = esq + N_EXP;                             // 8192  (80 KB total)

  rowsq_kernel<<<N_TOK + N_EXP, 256, 0, stream>>>(z, e, zsq, esq);

  dim3 g1(N_EXP / 64, N_TOK / 128);
  sim_gemm_kernel<<<g1, 256, 0, stream>>>(z, e, zsq, esq, sim);

  rowsum_kernel<<<N_TOK, 256, 0, stream>>>(sim, invrow);

  dim3 g2(DIM / 128, N_TOK / 32);
  weight_gemm_kernel<<<g2, 256, 0, stream>>>(sim, e, invrow, we);
}